// Translations_46127948759835
// MI455X (gfx1250) — compile-verified
//
#include <hip/hip_runtime.h>

// RBF vector-field apply: out[M,2] = exp(-||points_m - gd_n||^2 / sigma^2) @ controls
// M=16384, N=4096, DIM=2, sigma=1.
//
// Strategy (MI455X / gfx1250, wave32):
//   - Homogeneous trick: sq(m,n) = A_m . B_n with K=4:
//       A_m = s*[x0, x1, |x|^2, 1],  B_n = [-2*y0, -2*y1, 1, |y|^2],  s = -log2(e)/sigma^2
//     so one V_WMMA_F32_16X16X4_F32 yields D = -log2(e)*sq for a 16x16 tile.
//   - w = exp2(D) elementwise (v_exp_f32, co-executes with XDL WMMA).
//   - Contraction vs controls done per-lane (each lane owns one n column of D),
//     final 16-lane butterfly reduction per half-wave (matches D layout:
//     lanes 0-15 -> rows 0-7 in VGPRs 0-7, lanes 16-31 -> rows 8-15).

typedef __attribute__((ext_vector_type(2))) float v2f;
typedef __attribute__((ext_vector_type(8))) float v8f;

#define RBF_DIM    2
#define RBF_N      4096
#define RBF_M      16384
#define ROWS_PER_WAVE 16
#define WAVES_PER_BLOCK 8

__global__ __launch_bounds__(32 * WAVES_PER_BLOCK)
void rbf_translations_wmma(const float* __restrict__ gd,        // [N*2]
                           const float* __restrict__ controls,  // [N*2]
                           const float* __restrict__ points,    // [M*2]
                           float* __restrict__ out)             // [M*2]
{
    const int lane  = threadIdx.x & 31;
    const int wave  = blockIdx.x * WAVES_PER_BLOCK + (threadIdx.x >> 5);
    const int mBase = wave * ROWS_PER_WAVE;
    const int half  = lane >> 4;     // 0: lanes 0-15 (K=0,1 of A / rows 0-7 of D)
    const int sub   = lane & 15;     // row (A) / column n (B,D) within tile

    const float kNegLog2e = -1.4426950408889634f;  // s = -log2(e) / sigma^2, sigma=1

    // ---- A operand: 16x4 f32, 2 VGPRs/lane.
    // lanes 0-15: VGPR0=K0, VGPR1=K1 ; lanes 16-31: VGPR0=K2, VGPR1=K3 (row m = sub).
    const int m = mBase + sub;
    const float x0 = points[2 * m + 0];
    const float x1 = points[2 * m + 1];
    v2f a;
    if (half == 0) {
        a.x = kNegLog2e * x0;                 // K0
        a.y = kNegLog2e * x1;                 // K1
    } else {
        a.x = kNegLog2e * (x0 * x0 + x1 * x1);// K2
        a.y = kNegLog2e;                      // K3
    }

    float acc0[8], acc1[8];
#pragma unroll
    for (int r = 0; r < 8; ++r) { acc0[r] = 0.0f; acc1[r] = 0.0f; }

    for (int nBase = 0; nBase < RBF_N; nBase += 16) {
        const int n = nBase + sub;
        const float y0 = gd[2 * n + 0];
        const float y1 = gd[2 * n + 1];
        const float c0 = controls[2 * n + 0];
        const float c1 = controls[2 * n + 1];

        // ---- B operand: 4x16 f32, 2 VGPRs/lane (column n = sub).
        // lanes 0-15: VGPR0=K0, VGPR1=K1 ; lanes 16-31: VGPR0=K2, VGPR1=K3.
        v2f b;
        if (half == 0) {
            b.x = -2.0f * y0;            // K0
            b.y = -2.0f * y1;            // K1
        } else {
            b.x = 1.0f;                  // K2
            b.y = y0 * y0 + y1 * y1;     // K3
        }

        v8f czero = {};
        // D = A x B + 0 = -log2(e) * sqdist  for the 16x16 (m,n) tile
        v8f d = __builtin_amdgcn_wmma_f32_16x16x4_f32(
            /*neg_a=*/false, a, /*neg_b=*/false, b,
            /*c_mod=*/(short)0, czero, /*reuse_a=*/false, /*reuse_b=*/false);

        // w = 2^D = exp(-sqdist); accumulate w * controls[n,:] per lane.
#pragma unroll
        for (int r = 0; r < 8; ++r) {
            const float w = __builtin_amdgcn_exp2f(d[r]);
            acc0[r] = __builtin_fmaf(w, c0, acc0[r]);
            acc1[r] = __builtin_fmaf(w, c1, acc1[r]);
        }
    }

    // ---- Reduce over the 16 n-lanes within each half-wave (xor masks 1..8
    // never cross bit 4, so the two halves reduce independently).
#pragma unroll
    for (int r = 0; r < 8; ++r) {
#pragma unroll
        for (int off = 8; off >= 1; off >>= 1) {
            acc0[r] += __shfl_xor(acc0[r], off, 32);
            acc1[r] += __shfl_xor(acc1[r], off, 32);
        }
    }

    // lanes 0 and 16 hold complete sums: rows mBase+0..7 and mBase+8..15.
    if (sub == 0) {
        const int mRow = mBase + half * 8;
#pragma unroll
        for (int r = 0; r < 8; ++r) {
            out[2 * (mRow + r) + 0] = acc0[r];
            out[2 * (mRow + r) + 1] = acc1[r];
        }
    }
}

extern "C" void kernel_launch(void* const* d_in, const int* in_sizes, int n_in,
                              void* d_out, int out_size, void* d_ws, size_t ws_size,
                              hipStream_t stream) {
    (void)in_sizes; (void)n_in; (void)d_ws; (void)ws_size; (void)out_size;
    const float* gd       = (const float*)d_in[0];  // [N*2]
    const float* controls = (const float*)d_in[1];  // [N*2]
    const float* points   = (const float*)d_in[2];  // [M*2]
    float* out            = (float*)d_out;          // [M*2]

    const int totalWaves = RBF_M / ROWS_PER_WAVE;           // 1024
    const int blocks     = totalWaves / WAVES_PER_BLOCK;    // 128
    rbf_translations_wmma<<<blocks, 32 * WAVES_PER_BLOCK, 0, stream>>>(
        gd, controls, points, out);
}